// ssim_l1_loss_68908455297208
// MI455X (gfx1250) — compile-verified
//
#include <hip/hip_runtime.h>
#include <math.h>

typedef __attribute__((ext_vector_type(2))) float v2f;
typedef __attribute__((ext_vector_type(8))) float v8f;

// Problem constants (reference: (32,3,512,512), KSIZE=11, SIGMA=1.5)
static constexpr int   Himg   = 512;
static constexpr int   Wimg   = 512;
static constexpr int   BCimg  = 32 * 3;
static constexpr int   PADv   = 5;      // (11-1)/2
static constexpr int   TILE_H = 16;     // rows per wave tile
static constexpr int   STRIP_W= 64;     // cols per 4-wave block
static constexpr int   PROWS  = 28;     // 26 valid rows padded to K-mult-of-4
static constexpr int   PCOLS  = 80;     // 74 valid cols padded
static constexpr float C1c    = 0.0001f; // (0.01*1.0)^2
static constexpr float C2c    = 0.0009f; // (0.03*1.0)^2

__device__ __forceinline__ int reflect_idx(int i, int n) {
    i = (i < 0) ? -i : i;               // 'reflect' (no edge repeat); PAD < n
    i = (i >= n) ? (2 * n - 2 - i) : i;
    return i;
}

// Gaussian band weight: g[j] for j in [0,10], else 0 (window offset form)
__device__ __forceinline__ float gband(int j, float inv_sum) {
    if (j < 0 || j > 10) return 0.0f;
    float d = (float)j - 5.0f;
    return expf(-d * d / 4.5f) * inv_sum;   // 2*sigma^2 = 4.5
}

__global__ void ssim_l1_init_kernel(double* acc) {
    acc[0] = 0.0;
    acc[1] = 0.0;
}

__global__ void __launch_bounds__(128)
ssim_l1_main_kernel(const float* __restrict__ P, const float* __restrict__ T,
                    double* __restrict__ acc) {
    __shared__ float sP[PROWS][PCOLS];
    __shared__ float sT[PROWS][PCOLS];
    __shared__ float sI[4][16][34];     // per-wave vertical-blur intermediate (pad 34 vs bank conflicts)

    const int tid  = threadIdx.x;
    const int lane = tid & 31;
    const int wv   = tid >> 5;          // wave id 0..3 -> 16-col sub-tile
    const int r0   = blockIdx.y * TILE_H;
    const int c0   = blockIdx.x * STRIP_W;
    const size_t base = (size_t)blockIdx.z * Himg * Wimg;

    // ---- cooperative load of reflect-padded p,t strip (zero the K/N padding) ----
    for (int idx = tid; idx < PROWS * PCOLS; idx += 128) {
        const int row = idx / PCOLS;
        const int col = idx - row * PCOLS;
        float pv = 0.0f, tv = 0.0f;
        if (row < 26 && col < 74) {
            const int gr = reflect_idx(r0 - PADv + row, Himg);
            const int gc = reflect_idx(c0 - PADv + col, Wimg);
            const size_t g = base + (size_t)gr * Wimg + gc;
            pv = P[g];
            tv = T[g];
        }
        sP[row][col] = pv;
        sT[row][col] = tv;
    }
    __syncthreads();

    // ---- per-lane Gaussian band-matrix operand (same v2f serves as A for the
    //      vertical pass and as B for the horizontal pass; 16x16x4 f32 layout:
    //      lanes 0-15 carry K=4k+{0,1}, lanes 16-31 carry K=4k+{2,3}) ----
    float inv_sum;
    {
        float s = 0.0f;
        for (int j = 0; j < 11; ++j) { float d = (float)j - 5.0f; s += expf(-d * d / 4.5f); }
        inv_sum = 1.0f / s;
    }
    const int mn = lane & 15;           // M (vertical out-row) == N (horizontal out-col)
    const int hi = lane >> 4;
    v2f Wt[7];
    #pragma unroll
    for (int kc = 0; kc < 7; ++kc) {
        const int kb = 4 * kc + 2 * hi;
        Wt[kc].x = gband(kb - mn, inv_sum);
        Wt[kc].y = gband(kb + 1 - mn, inv_sum);
    }

    // ---- 5 stats: mu_p, mu_t, E[pp], E[tt], E[pt]; products made on the fly ----
    v8f res[5];
    #pragma unroll
    for (int s5 = 0; s5 < 5; ++s5) {
        // vertical blur: D = G(16x28) x S(28x16), two 16-col chunks of the halo
        #pragma unroll
        for (int nc = 0; nc < 2; ++nc) {
            v8f accv = {};
            const int col = 16 * wv + 16 * nc + mn;
            #pragma unroll
            for (int kc = 0; kc < 7; ++kc) {
                const int kr = 4 * kc + 2 * hi;
                const float p0 = sP[kr][col],     t0 = sT[kr][col];
                const float p1 = sP[kr + 1][col], t1 = sT[kr + 1][col];
                v2f b;
                switch (s5) {
                    case 0:  b.x = p0;      b.y = p1;      break;
                    case 1:  b.x = t0;      b.y = t1;      break;
                    case 2:  b.x = p0 * p0; b.y = p1 * p1; break;
                    case 3:  b.x = t0 * t0; b.y = t1 * t1; break;
                    default: b.x = p0 * t0; b.y = p1 * t1; break;
                }
                accv = __builtin_amdgcn_wmma_f32_16x16x4_f32(
                           false, Wt[kc], false, b, (short)0, accv, false, false);
            }
            // C/D layout: acc[i] = D[i + 8*hi][mn]
            #pragma unroll
            for (int i = 0; i < 8; ++i)
                sI[wv][i + 8 * hi][16 * nc + mn] = accv[i];
        }
        // horizontal blur: D = I(16x28) x Gt(28x16); same-wave LDS, in-order
        v8f acch = {};
        #pragma unroll
        for (int kc = 0; kc < 7; ++kc) {
            const int kb = 4 * kc + 2 * hi;
            v2f a;
            a.x = sI[wv][mn][kb];
            a.y = sI[wv][mn][kb + 1];
            acch = __builtin_amdgcn_wmma_f32_16x16x4_f32(
                       false, a, false, Wt[kc], (short)0, acch, false, false);
        }
        res[s5] = acch;
    }

    // ---- SSIM map + L1 over this wave's 16x16 tile ----
    float ssim_sum = 0.0f, l1_sum = 0.0f;
    #pragma unroll
    for (int i = 0; i < 8; ++i) {
        const float mu_p = res[0][i], mu_t = res[1][i];
        const float e_pp = res[2][i], e_tt = res[3][i], e_pt = res[4][i];
        const float sg_p  = e_pp - mu_p * mu_p;
        const float sg_t  = e_tt - mu_t * mu_t;
        const float sg_pt = e_pt - mu_p * mu_t;
        const float num = (2.0f * mu_p * mu_t + C1c) * (2.0f * sg_pt + C2c);
        const float den = (mu_p * mu_p + mu_t * mu_t + C1c) * (sg_p + sg_t + C2c);
        ssim_sum += num / den;
        const int M = i + 8 * hi;
        const float pv = sP[M + PADv][16 * wv + mn + PADv];
        const float tv = sT[M + PADv][16 * wv + mn + PADv];
        l1_sum += fabsf(pv - tv);
    }

    // ---- wave32 reduction then one double atomic per wave ----
    #pragma unroll
    for (int off = 16; off > 0; off >>= 1) {
        ssim_sum += __shfl_xor(ssim_sum, off, 32);
        l1_sum   += __shfl_xor(l1_sum,   off, 32);
    }
    if (lane == 0) {
        atomicAdd(&acc[0], (double)ssim_sum);
        atomicAdd(&acc[1], (double)l1_sum);
    }
}

__global__ void ssim_l1_final_kernel(const double* __restrict__ acc,
                                     float* __restrict__ out) {
    const double N = (double)BCimg * Himg * Wimg;   // 25,165,824
    const double ssim_mean = acc[0] / N;
    const double l1_mean   = acc[1] / N;
    out[0] = (float)(0.85 * (1.0 - ssim_mean) + 0.15 * l1_mean);
}

extern "C" void kernel_launch(void* const* d_in, const int* in_sizes, int n_in,
                              void* d_out, int out_size, void* d_ws, size_t ws_size,
                              hipStream_t stream) {
    const float* P = (const float*)d_in[0];
    const float* T = (const float*)d_in[1];
    double* acc = (double*)d_ws;
    float*  out = (float*)d_out;

    hipLaunchKernelGGL(ssim_l1_init_kernel, dim3(1), dim3(1), 0, stream, acc);

    dim3 grid(Wimg / STRIP_W, Himg / TILE_H, BCimg);   // (8, 32, 96)
    hipLaunchKernelGGL(ssim_l1_main_kernel, grid, dim3(128), 0, stream, P, T, acc);

    hipLaunchKernelGGL(ssim_l1_final_kernel, dim3(1), dim3(1), 0, stream, acc, out);
}